// NeuralGraphHidden_64682207477986
// MI455X (gfx1250) — compile-verified
//
#include <hip/hip_runtime.h>

// ---------------------------------------------------------------------------
// NeuralGraphHidden fused kernel for MI455X (gfx1250, wave32, WMMA).
// Grid: (M/16, B) workgroups x 256 threads (8 waves). One WG owns 16 atoms
// of one molecule and runs the full pipeline (message MLP, degree-sum,
// per-degree inner MLPs with degree masking) using V_WMMA_F32_16X16X4_F32.
// ---------------------------------------------------------------------------

#define BB    256
#define MM    128
#define DD    6
#define FA    128
#define FB    32
#define MSGW  128
#define CONVW 128
#define KIN   (FA + FB)     // 160
#define KSUM  (MSGW + FA)   // 256

// Padded LDS row strides (floats) to avoid 64-bank conflicts on A-fragment
// reads: 164%64=36, 132%64=4, 260%64=4 -> 16 rows hit 16 distinct banks.
#define SIN_LD  164
#define ST_LD   132
#define SSUM_LD 260

typedef __attribute__((ext_vector_type(2))) float v2f;
typedef __attribute__((ext_vector_type(8))) float v8f;

// Branchless ELU: one v_exp_f32 (TRANS op, co-issues with WMMA) + select.
// |exp(x)-1 - expm1(x)| ~ 1e-7 for x<=0: negligible vs reference tolerance.
__device__ __forceinline__ float eluf(float x) {
  float e = __expf(x) - 1.0f;
  return x > 0.0f ? x : e;
}

// One 16x16 output tile of A[16xK] * B[KxN-slice], fp32 WMMA, K%4==0.
// A in LDS (row-major, pitch lda); B global row-major (pitch ldb), offset to
// this tile's first column. Fragment layout per CDNA5 ISA 7.12.2:
//   A 16x4 : lane = M row (lane&15), VGPR{0,1} = K{kb,kb+1}, kb = 2*(lane>>4)
//   B 4x16 : lane = N col (lane&15), VGPR{0,1} = K{kb,kb+1}
//   C 16x16: VGPR r -> M = r + 8*(lane>>4), N = lane&15
template <int K>
__device__ __forceinline__ v8f wmma_gemm_tile(const float* __restrict__ A, int lda,
                                              const float* __restrict__ Bm, int ldb,
                                              int lane, v8f acc) {
  const int mr = lane & 15;
  const int kb = (lane >> 4) << 1;
  const float* ap = A + mr * lda + kb;
  const float* bp = Bm + kb * ldb + mr;
#pragma unroll 8
  for (int k = 0; k < K; k += 4) {
    v2f av, bv;
    av[0] = ap[0];
    av[1] = ap[1];
    bv[0] = bp[0];
    bv[1] = bp[ldb];
    acc = __builtin_amdgcn_wmma_f32_16x16x4_f32(
        /*neg_a=*/false, av, /*neg_b=*/false, bv,
        /*c_mod=*/(short)0, acc, /*reuse_a=*/false, /*reuse_b=*/false);
    ap += 4;
    bp += 4 * ldb;
  }
  return acc;
}

__global__ __launch_bounds__(256) void ngh_fused_kernel(
    const float* __restrict__ atoms,     // (B, M, FA)
    const float* __restrict__ bonds,     // (B, M, D, FB)
    const int* __restrict__ edges,       // (B, M, D)
    const float* __restrict__ msg_w0,    // (KIN, MSGW)
    const float* __restrict__ msg_w1,    // (MSGW, MSGW)
    const float* __restrict__ inner_w0,  // (D, KSUM, CONVW)
    const float* __restrict__ inner_w1,  // (D, CONVW, CONVW)
    float* __restrict__ out) {           // (B, M, CONVW)
  __shared__ float s_in[16 * SIN_LD];    // msg_in tile: 16 rows x 160 (padded)
  __shared__ float s_t[16 * ST_LD];      // inter-layer activation tile 16x128
  __shared__ float s_sum[16 * SSUM_LD];  // [summed_msg(128) | atoms(128)] per atom
  __shared__ float s_out[16 * CONVW];    // output accumulator 16x128
  __shared__ int s_edge[96];             // 16 atoms x 6 edges
  __shared__ int s_deg[16];              // atom degrees

  const int tid = threadIdx.x;
  const int wave = tid >> 5;   // 0..7 (wave32)
  const int lane = tid & 31;
  const int b = blockIdx.y;    // molecule
  const int mt = blockIdx.x;   // atom tile
  const int m0 = mt * 16;

  // Warm L2/WGP$ for the stage-A weights (global_prefetch_b8).
  for (int off = tid * 32; off < KIN * MSGW; off += 256 * 32)
    __builtin_prefetch(msg_w0 + off, 0, 3);
  for (int off = tid * 32; off < MSGW * MSGW; off += 256 * 32)
    __builtin_prefetch(msg_w1 + off, 0, 3);

  if (tid < 96)
    s_edge[tid] = edges[((size_t)b * MM + m0 + tid / 6) * DD + tid % 6];

  // Async copy (CDNA5 ASYNCcnt path) of the atoms half of `summed` into LDS;
  // it is only consumed in stage B, but it is waited on (s_wait_asynccnt 0) before stage A.
  for (int i = tid; i < 16 * 32; i += 256) {  // 512 x float4
    const int ml = i >> 5;                    // 32 float4 per row
    const int c4 = (i & 31) << 2;
    unsigned lds_off = (unsigned)(uintptr_t)&s_sum[ml * SSUM_LD + MSGW + c4];
    unsigned long long ga =
        (unsigned long long)(atoms + ((size_t)b * MM + m0 + ml) * FA + c4);
    asm volatile("global_load_async_to_lds_b128 %0, %1, off"
                 :: "v"(lds_off), "v"(ga)
                 : "memory");
  }

  // Zero the msg-sum half of s_sum and the output accumulator.
  for (int i = tid; i < 16 * 128; i += 256) {
    const int ml = i >> 7, c = i & 127;
    s_sum[ml * SSUM_LD + c] = 0.0f;
    s_out[i] = 0.0f;
  }
  asm volatile("s_wait_asynccnt 0x0" ::: "memory");
  __syncthreads();

  if (tid < 16) {
    int cnt = 0;
#pragma unroll
    for (int d = 0; d < DD; ++d) cnt += (s_edge[tid * DD + d] >= 0) ? 1 : 0;
    s_deg[tid] = cnt;  // deg==6 never matches the 0..5 mask -> output 0 (as ref)
  }

  // ============== Stage A: message MLP over 96 (atom,deg) rows ==============
  for (int rt = 0; rt < 6; ++rt) {
    // Gather msg_in rows rt*16 .. rt*16+15: [neighbour atoms(128) | bonds(32)]
    for (int idx = tid; idx < 16 * 40; idx += 256) {
      const int rr = idx / 40;  // row in tile
      const int q = idx % 40;   // float4 slot; col = q*4 (0..159)
      const int row = rt * 16 + rr;
      const int ml = row / 6, d = row % 6;
      float4 v;
      if (q < 32) {
        const int e = s_edge[row];
        v = (e >= 0)
                ? *(const float4*)(atoms + ((size_t)b * MM + e) * FA + q * 4)
                : make_float4(0.f, 0.f, 0.f, 0.f);
      } else {
        v = *(const float4*)(bonds +
                             (((size_t)b * MM + m0 + ml) * DD + d) * FB +
                             (q - 32) * 4);
      }
      *(float4*)&s_in[rr * SIN_LD + q * 4] = v;
    }
    __syncthreads();

    {  // layer 0: 16x160 @ 160x128, wave w owns column tile w
      v8f acc = {};
      acc = wmma_gemm_tile<KIN>(s_in, SIN_LD, msg_w0 + wave * 16, MSGW, lane, acc);
      const int n = lane & 15, rb = (lane >> 4) << 3;
#pragma unroll
      for (int r = 0; r < 8; ++r)
        s_t[(rb + r) * ST_LD + wave * 16 + n] = eluf(acc[r]);
    }
    __syncthreads();

    {  // layer 1: 16x128 @ 128x128, then elu + per-atom degree-sum (ds_add_f32)
      v8f acc = {};
      acc = wmma_gemm_tile<MSGW>(s_t, ST_LD, msg_w1 + wave * 16, MSGW, lane, acc);
      const int n = lane & 15, rb = (lane >> 4) << 3;
#pragma unroll
      for (int r = 0; r < 8; ++r) {
        const int row = rt * 16 + rb + r;
        const int ml = row / 6;
        atomicAdd(&s_sum[ml * SSUM_LD + wave * 16 + n], eluf(acc[r]));
      }
    }
    __syncthreads();
  }

  // ====== Stage B: per-degree inner MLPs on summed = [msg_sum | atoms] ======
  int rdeg[8];  // degree of each output row this lane owns (hoisted)
  {
    const int rb = (lane >> 4) << 3;
#pragma unroll
    for (int r = 0; r < 8; ++r) rdeg[r] = s_deg[rb + r];
  }

  for (int deg = 0; deg < DD; ++deg) {
    {  // inner layer 0: 16x256 @ 256x128
      v8f acc = {};
      acc = wmma_gemm_tile<KSUM>(
          s_sum, SSUM_LD,
          inner_w0 + (size_t)deg * KSUM * CONVW + wave * 16, CONVW, lane, acc);
      const int n = lane & 15, rb = (lane >> 4) << 3;
#pragma unroll
      for (int r = 0; r < 8; ++r)
        s_t[(rb + r) * ST_LD + wave * 16 + n] = eluf(acc[r]);
    }
    __syncthreads();

    {  // inner layer 1: 16x128 @ 128x128, elu, masked accumulate by degree
      v8f acc = {};
      acc = wmma_gemm_tile<CONVW>(
          s_t, ST_LD,
          inner_w1 + (size_t)deg * CONVW * CONVW + wave * 16, CONVW, lane, acc);
      const int n = lane & 15, rb = (lane >> 4) << 3;
#pragma unroll
      for (int r = 0; r < 8; ++r) {
        if (rdeg[r] == deg)
          s_out[(rb + r) * CONVW + wave * 16 + n] += eluf(acc[r]);
      }
    }
    __syncthreads();
  }

  // ---- write 16x128 output tile ----
  for (int i = tid; i < 16 * CONVW; i += 256) {
    const int ml = i >> 7, c = i & 127;
    out[((size_t)b * MM + m0 + ml) * CONVW + c] = s_out[i];
  }
}

// ---------------------------------------------------------------------------
extern "C" void kernel_launch(void* const* d_in, const int* in_sizes, int n_in,
                              void* d_out, int out_size, void* d_ws,
                              size_t ws_size, hipStream_t stream) {
  const float* atoms = (const float*)d_in[0];
  const float* bonds = (const float*)d_in[1];
  const int* edges = (const int*)d_in[2];
  const float* msg_w0 = (const float*)d_in[3];
  const float* msg_w1 = (const float*)d_in[4];
  const float* inner_w0 = (const float*)d_in[5];
  const float* inner_w1 = (const float*)d_in[6];
  float* out = (float*)d_out;

  dim3 grid(MM / 16, BB);  // (8, 256) workgroups
  ngh_fused_kernel<<<grid, 256, 0, stream>>>(atoms, bonds, edges, msg_w0,
                                             msg_w1, inner_w0, inner_w1, out);
  (void)in_sizes; (void)n_in; (void)out_size; (void)d_ws; (void)ws_size;
}